// Cluster_loss_TwoView_91276644974681
// MI455X (gfx1250) — compile-verified
//
#include <hip/hip_runtime.h>
#include <hip/hip_bf16.h>

// ---------------------------------------------------------------------------
// Cluster loss, two-view.  Bandwidth-bound streaming pass (1.02 GB ~ 44 us at
// 23.3 TB/s) + tiny WMMA Gram-diagonal finalize on the 1008x128 accumulator.
// ---------------------------------------------------------------------------

#define MARGIN       0.1f
#define NUM_CLASSES  1000
#define DIMS         128
#define PAD_CLASSES  1008          // 63 tiles of 16 classes (1000..1007 stay zero)
#define NUM_TILES    (PAD_CLASSES / 16)

typedef __attribute__((ext_vector_type(2))) float v2f;
typedef __attribute__((ext_vector_type(8))) float v8f;

// ---- Kernel 1: zero accumulator, counts, and the scalar output ------------
__global__ void cl_zero_kernel(float* __restrict__ acc_and_counts, int nelems,
                               float* __restrict__ out) {
    int i = blockIdx.x * blockDim.x + threadIdx.x;
    int stride = gridDim.x * blockDim.x;
    for (int j = i; j < nelems; j += stride) acc_and_counts[j] = 0.0f;
    if (i == 0) out[0] = 0.0f;
}

// ---- Kernel 2: streaming pass. One wave32 per row. ------------------------
// lane loads float4 (coalesced b128), wave-reduce both squared norms,
// scatter normalized difference into acc[label][*] with fp32 global atomics.
__global__ void cl_accum_kernel(const float* __restrict__ f1,
                                const float* __restrict__ f2,
                                const int*   __restrict__ label,
                                float* __restrict__ acc,
                                float* __restrict__ counts,
                                int n_rows) {
    const int lane   = threadIdx.x & 31;
    const int wave   = (blockIdx.x * blockDim.x + threadIdx.x) >> 5;
    const int nwaves = (gridDim.x * blockDim.x) >> 5;

    for (int row = wave; row < n_rows; row += nwaves) {
        const size_t base = (size_t)row * DIMS + (size_t)lane * 4;
        const float4 a = *(const float4*)(f1 + base);
        const float4 b = *(const float4*)(f2 + base);

        float s1 = a.x * a.x + a.y * a.y + a.z * a.z + a.w * a.w;
        float s2 = b.x * b.x + b.y * b.y + b.z * b.z + b.w * b.w;
        // wave32 butterfly reduction for both norms
        #pragma unroll
        for (int off = 16; off > 0; off >>= 1) {
            s1 += __shfl_xor(s1, off, 32);
            s2 += __shfl_xor(s2, off, 32);
        }
        const float r1 = 1.0f / sqrtf(s1);
        const float r2 = 1.0f / sqrtf(s2);

        const int lbl = label[row];
        float* dst = acc + (size_t)lbl * DIMS + (size_t)lane * 4;
        unsafeAtomicAdd(dst + 0, a.x * r1 - b.x * r2);
        unsafeAtomicAdd(dst + 1, a.y * r1 - b.y * r2);
        unsafeAtomicAdd(dst + 2, a.z * r1 - b.z * r2);
        unsafeAtomicAdd(dst + 3, a.w * r1 - b.w * r2);
        if (lane == 0) unsafeAtomicAdd(counts + lbl, 1.0f);
    }
}

// ---- Kernel 3: WMMA finalize. One wave per 16-class tile. -----------------
// per_class[c] = sum_k (acc[c][k]/n_c)^2  ==  diag of G*G^T for the tile.
// For V_WMMA_F32_16X16X4_F32 the f32 A-layout and the transposed B-layout
// coincide register-for-register, so the same 2-VGPR operand serves as both
// A and B.  32 WMMA issues accumulate K=128.
__global__ void cl_finalize_kernel(const float* __restrict__ acc,
                                   const float* __restrict__ counts,
                                   float* __restrict__ out) {
    const int tile  = blockIdx.x;            // 0..NUM_TILES-1
    const int lane  = threadIdx.x;           // 0..31, all lanes active (EXEC all-1s)
    const int r     = lane & 15;             // row (A) == col (B) index in tile
    const int c     = tile * 16 + r;         // class this lane carries
    const float cnt = counts[c];
    const float inv = 1.0f / fmaxf(cnt, 1.0f);
    const float* gb = acc + (size_t)c * DIMS;
    const int khalf = (lane >> 4) * 2;       // lanes 0-15: K pair {0,1}; 16-31: {2,3}

    v8f d = {};
    #pragma unroll
    for (int k0 = 0; k0 < DIMS; k0 += 4) {
        v2f ab;
        ab.x = gb[k0 + khalf + 0] * inv;
        ab.y = gb[k0 + khalf + 1] * inv;
        // D += A * A^T  (same operand is simultaneously the A and B layout)
        d = __builtin_amdgcn_wmma_f32_16x16x4_f32(
                /*neg_a=*/false, ab, /*neg_b=*/false, ab,
                /*c_mod=*/(short)0, d, /*reuse_a=*/false, /*reuse_b=*/false);
    }

    // Diagonal ownership: lanes 0-7 hold D[m][m] in vgpr m (m = lane);
    // lanes 24-31 hold D[m][m] in vgpr m-8 (m = lane-16).
    const bool valid = (lane < 8) || (lane >= 24);
    const int  idx   = (lane < 8) ? lane : (lane - 24);
    float diag = d[0];
    if (idx == 1) diag = d[1];
    if (idx == 2) diag = d[2];
    if (idx == 3) diag = d[3];
    if (idx == 4) diag = d[4];
    if (idx == 5) diag = d[5];
    if (idx == 6) diag = d[6];
    if (idx == 7) diag = d[7];

    float contrib = 0.0f;
    if (valid && cnt > 0.0f) contrib = fmaxf(diag - MARGIN, 0.0f);

    #pragma unroll
    for (int off = 16; off > 0; off >>= 1)
        contrib += __shfl_xor(contrib, off, 32);
    if (lane == 0) unsafeAtomicAdd(out, contrib);
}

// ---------------------------------------------------------------------------
extern "C" void kernel_launch(void* const* d_in, const int* in_sizes, int n_in,
                              void* d_out, int out_size, void* d_ws, size_t ws_size,
                              hipStream_t stream) {
    const float* f1    = (const float*)d_in[0];
    const float* f2    = (const float*)d_in[1];
    const int*   label = (const int*)d_in[2];
    float*       out   = (float*)d_out;

    const int n_rows = in_sizes[0] / DIMS;

    float* acc    = (float*)d_ws;                       // [PAD_CLASSES][DIMS]
    float* counts = acc + (size_t)PAD_CLASSES * DIMS;   // [PAD_CLASSES]
    const int nz  = PAD_CLASSES * DIMS + PAD_CLASSES;

    cl_zero_kernel<<<(nz + 255) / 256, 256, 0, stream>>>(acc, nz, out);

    const int threads = 256;                 // 8 wave32 per block
    const int blocks  = 4096;                // 32768 waves, grid-stride over rows
    cl_accum_kernel<<<blocks, threads, 0, stream>>>(f1, f2, label, acc, counts, n_rows);

    cl_finalize_kernel<<<NUM_TILES, 32, 0, stream>>>(acc, counts, out);
}